// SelfAttentionWithMap_63273458204821
// MI455X (gfx1250) — compile-verified
//
#include <hip/hip_runtime.h>

// Self-attention with map, B=C=1, N=16384 (128x128), NO=1.
// Rank-1 score matrix => never materialize NxN. Two N^2 exp passes with
// reductions done on the matrix unit via V_WMMA_F32_16X16X4_F32.
// Branch-free fragments; log2(e) folded into loop-invariant operand and raw
// v_exp_f32 via __builtin_amdgcn_exp2f => each exp is one v_mul + v_exp_f32.

typedef __attribute__((ext_vector_type(2))) float v2f;
typedef __attribute__((ext_vector_type(8))) float v8f;

#define NPIX 16384
#define CHUNK 2048
#define WAVES_PER_BLOCK 8
#define LOG2E 1.44269504088896340736f

// ---------------------------------------------------------------- kernel 1 --
// f[n] = sign(wq)*x[n]+bq ; g[n] = sign(wk)*x[n]+bk ; hv[n] = sign(wv)*x[n]+bv
__global__ void fgh_kernel(const float* __restrict__ x,
                           const float* __restrict__ wq, const float* __restrict__ bq,
                           const float* __restrict__ wk, const float* __restrict__ bk,
                           const float* __restrict__ wv, const float* __restrict__ bv,
                           float* __restrict__ f, float* __restrict__ g,
                           float* __restrict__ hv)
{
    int i = blockIdx.x * blockDim.x + threadIdx.x;
    if (i >= NPIX) return;
    float sq = wq[0] / fabsf(wq[0]);   // spectral norm of 1x1 weight = |w|
    float sk = wk[0] / fabsf(wk[0]);
    float sv = wv[0] / fabsf(wv[0]);
    float xv = x[i];
    f[i]  = sq * xv + bq[0];
    g[i]  = sk * xv + bk[0];
    hv[i] = sv * xv + bv[0];
}

// ---------------------------------------------------------------- kernel 2 --
// Per wave: 16 columns j. Accumulate over i:
//   S0[j] = sum_i exp(f[i]*g[j]),  S1[j] = sum_i hv[i]*exp(f[i]*g[j])
// D = A x B + C with A(16x4): row0 = ones, row1 = hv chunk, rest 0;
// B(4x16): E[k, j] = exp2(f * (g*log2e)).
// Then o[j] = gamma*S1/S0 + x[j];  R[j] = 1/(N*S0[j]) for the row pass.
__global__ __launch_bounds__(256) void col_kernel(
    const float* __restrict__ f, const float* __restrict__ g,
    const float* __restrict__ hv, const float* __restrict__ x,
    const float* __restrict__ gamma,
    float* __restrict__ o, float* __restrict__ R)
{
    __shared__ float sf[CHUNK];
    __shared__ float sh[CHUNK];

    const int tid  = threadIdx.x;
    const int lane = tid & 31;
    const int wave = tid >> 5;
    const int tile = blockIdx.x * WAVES_PER_BLOCK + wave; // 16 columns per tile
    const int j0   = tile * 16;
    const int col  = lane & 15;          // N index (and A-row index m)
    const int kh   = (lane >> 4) * 2;    // which K pair this lane holds

    const float gcl = g[j0 + col] * LOG2E;   // exp(f*g) = exp2(f*gcl)
    // Branch-free A construction: A = selOne*1 + selH*hv
    const float selOne = (col == 0) ? 1.0f : 0.0f;
    const float selH   = (col == 1) ? 1.0f : 0.0f;

    v8f acc = {0.0f, 0.0f, 0.0f, 0.0f, 0.0f, 0.0f, 0.0f, 0.0f};

    for (int base = 0; base < NPIX; base += CHUNK) {
        __syncthreads();
        for (int t = tid; t < CHUNK; t += 256) {
            sf[t] = f[base + t];
            sh[t] = hv[base + t];
        }
        __syncthreads();
        #pragma unroll 8
        for (int k = 0; k < CHUNK; k += 4) {
            float f0 = sf[k + kh];
            float f1 = sf[k + kh + 1];
            float h0 = sh[k + kh];
            float h1 = sh[k + kh + 1];
            v2f B;
            B.x = __builtin_amdgcn_exp2f(f0 * gcl);  // raw v_exp_f32, no fixup
            B.y = __builtin_amdgcn_exp2f(f1 * gcl);
            v2f A;
            A.x = selOne + selH * h0;    // fma, no divergence
            A.y = selOne + selH * h1;
            acc = __builtin_amdgcn_wmma_f32_16x16x4_f32(
                false, A, false, B, (short)0, acc, false, false);
        }
    }
    // D layout: VGPR v, lanes 0-15 -> (M=v, N=lane). Row 0 = S0, row 1 = S1.
    if (lane < 16) {
        float S0 = acc[0];
        float S1 = acc[1];
        o[j0 + col] = gamma[0] * (S1 / S0) + x[j0 + col];
        R[j0 + col] = 1.0f / ((float)NPIX * S0);   // 1/N folded in for row pass
    }
}

// ---------------------------------------------------------------- kernel 3 --
// Per wave: 16 rows i. attn[i] = sum_j exp(f[i]*g[j]) * R[j]   (R has 1/N)
// A(16x4): E[i, k] = exp2((f[i]*log2e)*g[jc+k]); B(4x16): col0 = R chunk.
__global__ __launch_bounds__(256) void row_kernel(
    const float* __restrict__ f, const float* __restrict__ g,
    const float* __restrict__ R, float* __restrict__ attn)
{
    __shared__ float sg[CHUNK];
    __shared__ float sr[CHUNK];

    const int tid  = threadIdx.x;
    const int lane = tid & 31;
    const int wave = tid >> 5;
    const int tile = blockIdx.x * WAVES_PER_BLOCK + wave; // 16 rows per tile
    const int i0   = tile * 16;
    const int m    = lane & 15;          // A-row index (and B column index)
    const int kh   = (lane >> 4) * 2;

    const float frl  = f[i0 + m] * LOG2E;        // loop-invariant, log2e folded
    const float selR = (m == 0) ? 1.0f : 0.0f;   // branch-free B construction

    v8f acc = {0.0f, 0.0f, 0.0f, 0.0f, 0.0f, 0.0f, 0.0f, 0.0f};

    for (int base = 0; base < NPIX; base += CHUNK) {
        __syncthreads();
        for (int t = tid; t < CHUNK; t += 256) {
            sg[t] = g[base + t];
            sr[t] = R[base + t];
        }
        __syncthreads();
        #pragma unroll 8
        for (int k = 0; k < CHUNK; k += 4) {
            float g0 = sg[k + kh];
            float g1 = sg[k + kh + 1];
            float r0 = sr[k + kh];
            float r1 = sr[k + kh + 1];
            v2f A;
            A.x = __builtin_amdgcn_exp2f(frl * g0);  // raw v_exp_f32, no fixup
            A.y = __builtin_amdgcn_exp2f(frl * g1);
            v2f B;
            B.x = selR * r0;             // mul, no divergence
            B.y = selR * r1;
            acc = __builtin_amdgcn_wmma_f32_16x16x4_f32(
                false, A, false, B, (short)0, acc, false, false);
        }
    }
    // Column N=0 of D: lane 0 holds M=0..7 in acc[0..7]; lane 16 holds M=8..15.
    if (m == 0) {
        int mb = (lane >> 4) * 8;
        #pragma unroll
        for (int v = 0; v < 8; ++v)
            attn[i0 + mb + v] = acc[v];
    }
}

// --------------------------------------------------------------------------
extern "C" void kernel_launch(void* const* d_in, const int* in_sizes, int n_in,
                              void* d_out, int out_size, void* d_ws, size_t ws_size,
                              hipStream_t stream)
{
    (void)in_sizes; (void)n_in; (void)out_size; (void)ws_size;
    const float* x     = (const float*)d_in[0];
    const float* wq    = (const float*)d_in[1];
    const float* bq    = (const float*)d_in[2];
    const float* wk    = (const float*)d_in[3];
    const float* bk    = (const float*)d_in[4];
    const float* wv    = (const float*)d_in[5];
    const float* bv    = (const float*)d_in[6];
    const float* gamma = (const float*)d_in[7];

    float* out = (float*)d_out;          // [0,N) = o ; [N,2N) = attn_map
    float* f   = (float*)d_ws;           // ws: f | g | hv | R  (4 * 64KB)
    float* g   = f + NPIX;
    float* hv  = g + NPIX;
    float* R   = hv + NPIX;

    fgh_kernel<<<NPIX / 256, 256, 0, stream>>>(x, wq, bq, wk, bk, wv, bv, f, g, hv);

    const int tiles = NPIX / 16;                       // 1024 wave-tiles
    const int blocks = tiles / WAVES_PER_BLOCK;        // 128 blocks of 256 thr
    col_kernel<<<blocks, 256, 0, stream>>>(f, g, hv, x, gamma, out, R);
    row_kernel<<<blocks, 256, 0, stream>>>(f, g, R, out + NPIX);
}